// QMultiTask_12506944766104
// MI455X (gfx1250) — compile-verified
//
#include <hip/hip_runtime.h>
#include <math.h>

// ---------------------------------------------------------------------------
// QMultiTask on MI455X (gfx1250, wave32, WMMA).
// B=16, T=40, D=128, CELL=64, NE=6, NA=12, dims={300,74,35}
// ---------------------------------------------------------------------------

#define LSTR 132   // padded LDS row stride (floats) -> bank spread
#define TT 40

typedef __attribute__((ext_vector_type(2))) float v2f;
typedef __attribute__((ext_vector_type(8))) float v8f;

// ---------------------------------------------------------------------------
// 128x128x128 fp32 matmul on one 256-thread block (8 waves) using
// V_WMMA_F32_16X16X4_F32, register-blocked 4x2 tiles per wave:
// per K-step: 4 A-frags + 2 B-frags feed 8 WMMAs (0.75 LDS loads / WMMA).
// BMODE: 0 = B from LDS (row-major, stride LSTR)
//        1 = B from LDS, transposed source (B = S^T), stride LSTR
//        2 = B from global (row-major, stride 128)
// EXTRA: 0 none
//        1 C += s*(u[r]*u[c] + w[r]*w[c])   (Hermitian rank-2, real part)
//        2 C += s*(u[r]*w[c] - w[r]*u[c])   (antisymmetric, imag part)
// ---------------------------------------------------------------------------
template <int BMODE, int EXTRA>
__device__ __forceinline__ void mm128(float* __restrict__ Cl,
                                      const float* __restrict__ Al,
                                      const float* __restrict__ Bsrc,
                                      float scale, float s,
                                      const float* __restrict__ uvec,
                                      const float* __restrict__ wvec,
                                      int tid) {
  const int wave = tid >> 5;
  const int lane = tid & 31;
  const int l = lane & 15;   // row (A) / col (B,C)
  const int h = lane >> 4;   // half-wave: carries K pairs (A/B), M+8 (C/D)
  // wave -> 4x2 block of 16x16 tiles: rows [bm*64, bm*64+64), cols [bn*32, bn*32+32)
  const int bm = wave >> 2;        // 0..1
  const int bn = wave & 3;         // 0..3
  const int m0 = bm * 64;
  const int n0 = bn * 32;

  v8f acc[4][2];
#pragma unroll
  for (int i = 0; i < 4; ++i)
#pragma unroll
    for (int j = 0; j < 2; ++j) acc[i][j] = {};

#pragma unroll 2
  for (int k0 = 0; k0 < 128; k0 += 4) {
    v2f a[4], bfr[2];
#pragma unroll
    for (int i = 0; i < 4; ++i) {
      const float* p = Al + (m0 + 16 * i + l) * LSTR + k0 + 2 * h;
      a[i].x = p[0];
      a[i].y = p[1];
    }
#pragma unroll
    for (int j = 0; j < 2; ++j) {
      if (BMODE == 0) {
        bfr[j].x = Bsrc[(k0 + 2 * h) * LSTR + n0 + 16 * j + l];
        bfr[j].y = Bsrc[(k0 + 2 * h + 1) * LSTR + n0 + 16 * j + l];
      } else if (BMODE == 1) {
        const float* p = Bsrc + (n0 + 16 * j + l) * LSTR + k0 + 2 * h;
        bfr[j].x = p[0];
        bfr[j].y = p[1];
      } else {
        bfr[j].x = Bsrc[(k0 + 2 * h) * 128 + n0 + 16 * j + l];
        bfr[j].y = Bsrc[(k0 + 2 * h + 1) * 128 + n0 + 16 * j + l];
      }
    }
#pragma unroll
    for (int i = 0; i < 4; ++i)
#pragma unroll
      for (int j = 0; j < 2; ++j)
        acc[i][j] = __builtin_amdgcn_wmma_f32_16x16x4_f32(
            false, a[i], false, bfr[j], (short)0, acc[i][j], false, false);
  }

#pragma unroll
  for (int i = 0; i < 4; ++i) {
#pragma unroll
    for (int j = 0; j < 2; ++j) {
#pragma unroll
      for (int r = 0; r < 8; ++r) {
        const int row = m0 + 16 * i + r + 8 * h;
        const int col = n0 + 16 * j + l;
        float v = acc[i][j][r] * scale;
        if (EXTRA == 1) v += s * (uvec[row] * uvec[col] + wvec[row] * wvec[col]);
        if (EXTRA == 2) v += s * (uvec[row] * wvec[col] - wvec[row] * uvec[col]);
        Cl[row * LSTR + col] = v;
      }
    }
  }
}

// ---------------------------------------------------------------------------
// Kernel 1: projection + relu + norm + phase; emit a = Ux*pr, b = Ux*pi.
// grid = 3*640 blocks of 128 threads.
// ---------------------------------------------------------------------------
__global__ void prep_kernel(const float* __restrict__ xt,
                            const float* __restrict__ xa,
                            const float* __restrict__ xv,
                            const float* __restrict__ smask,
                            const float* __restrict__ Wpt, const float* __restrict__ bpt,
                            const float* __restrict__ Uxt,
                            const float* __restrict__ Wpa, const float* __restrict__ bpa,
                            const float* __restrict__ Uxa,
                            const float* __restrict__ Wpv, const float* __restrict__ bpv,
                            const float* __restrict__ Uxv,
                            const float* __restrict__ phase_tab,
                            float* __restrict__ Ag, float* __restrict__ Bg,
                            float* __restrict__ norms) {
  const int m = blockIdx.x / 640;
  const int i = blockIdx.x % 640;  // b*40+t
  const int tid = threadIdx.x;     // 0..127
  const float* x = (m == 0) ? xt : (m == 1) ? xa : xv;
  const float* Wp = (m == 0) ? Wpt : (m == 1) ? Wpa : Wpv;
  const float* bp = (m == 0) ? bpt : (m == 1) ? bpa : bpv;
  const float* Ux = (m == 0) ? Uxt : (m == 1) ? Uxa : Uxv;
  const int dim = (m == 0) ? 300 : (m == 1) ? 74 : 35;

  __shared__ float sh[128];
  __shared__ float prs[128];
  __shared__ float pis[128];

  float acc = bp[tid];
  const float* xr = x + i * dim;
  for (int j = 0; j < dim; ++j) acc += xr[j] * Wp[j * 128 + tid];
  const float rep = fmaxf(acc, 0.0f);

  sh[tid] = rep * rep;
  __syncthreads();
  for (int off = 64; off > 0; off >>= 1) {
    if (tid < off) sh[tid] += sh[tid + off];
    __syncthreads();
  }
  const float norm = sqrtf(sh[0]);
  if (tid == 0) norms[m * 640 + i] = norm;
  const float amp = rep / fmaxf(norm, 1e-12f);

  const int spk = (smask[i * 2 + 1] > smask[i * 2 + 0]) ? 1 : 0;
  const float ph = phase_tab[spk * 128 + tid];
  prs[tid] = amp * cosf(ph);
  pis[tid] = amp * sinf(ph);
  __syncthreads();

  float sa = 0.0f, sb = 0.0f;
  const float* urow = Ux + tid * 128;
  for (int j = 0; j < 128; ++j) {
    const float u = urow[j];
    sa += u * prs[j];
    sb += u * pis[j];
  }
  Ag[(m * 640 + i) * 128 + tid] = sa;
  Bg[(m * 640 + i) * 128 + tid] = sb;
}

// ---------------------------------------------------------------------------
// Kernel 2: softmax weights over modality norms.
// ---------------------------------------------------------------------------
__global__ void wts_kernel(const float* __restrict__ norms, float* __restrict__ wts) {
  const int i = blockIdx.x * blockDim.x + threadIdx.x;
  if (i >= 640) return;
  const float n0 = norms[i], n1 = norms[640 + i], n2 = norms[1280 + i];
  const float mx = fmaxf(n0, fmaxf(n1, n2));
  const float e0 = expf(n0 - mx), e1 = expf(n1 - mx), e2 = expf(n2 - mx);
  const float inv = 1.0f / (e0 + e1 + e2);
  wts[i * 3 + 0] = e0 * inv;
  wts[i * 3 + 1] = e1 * inv;
  wts[i * 3 + 2] = e2 * inv;
}

// ---------------------------------------------------------------------------
// Kernel 3: L2-normalize measurement rows of K_emo/K_act; emit row-major
// copies (for diag reductions) and transposed copies (coalesced WMMA B).
// grid = 256 blocks (e*128+k) of 128 threads (d).
// ---------------------------------------------------------------------------
__global__ void normk_kernel(const float* __restrict__ Kemo,
                             const float* __restrict__ Kact,
                             float* __restrict__ VrE, float* __restrict__ ViE,
                             float* __restrict__ VrA, float* __restrict__ ViA,
                             float* __restrict__ VTrE, float* __restrict__ VTiE,
                             float* __restrict__ VTrA, float* __restrict__ VTiA) {
  const int e = blockIdx.x >> 7;
  const int k = blockIdx.x & 127;
  const int d = threadIdx.x;
  const float* K = e ? Kact : Kemo;
  const float kr = K[(k * 128 + d) * 2 + 0];
  const float ki = K[(k * 128 + d) * 2 + 1];
  __shared__ float sh[128];
  sh[d] = kr * kr + ki * ki;
  __syncthreads();
  for (int off = 64; off > 0; off >>= 1) {
    if (d < off) sh[d] += sh[d + off];
    __syncthreads();
  }
  const float nrm = sqrtf(sh[0] + 1e-12f);
  const float vr = kr / nrm, vi = ki / nrm;
  float* Vr = e ? VrA : VrE;
  float* Vi = e ? ViA : ViE;
  float* VTr = e ? VTrA : VTrE;
  float* VTi = e ? VTiA : VTiE;
  Vr[k * 128 + d] = vr;
  Vi[k * 128 + d] = vi;
  VTr[d * 128 + k] = vr;
  VTi[d * 128 + k] = vi;
}

// ---------------------------------------------------------------------------
// Kernel 4: QRNN density-matrix scan + per-modality measurement.
// One block per (modality, batch): 48 blocks x 256 threads (8 waves).
// Dynamic LDS: Uh, Hr, Hi, Tmp (128x132 each) + a/b vectors + reduce buffer.
// ---------------------------------------------------------------------------
__global__ void qrnn_kernel(const float* __restrict__ Uht,
                            const float* __restrict__ Uha,
                            const float* __restrict__ Uhv,
                            const float* __restrict__ LamT,
                            const float* __restrict__ LamA,
                            const float* __restrict__ LamV,
                            const float* __restrict__ Ag,
                            const float* __restrict__ Bg,
                            const float* __restrict__ VTrE, const float* __restrict__ VTiE,
                            const float* __restrict__ VTrA, const float* __restrict__ VTiA,
                            const float* __restrict__ VrE, const float* __restrict__ ViE,
                            const float* __restrict__ VrA, const float* __restrict__ ViA,
                            float* __restrict__ Pem, float* __restrict__ Pam) {
  extern __shared__ float smem[];
  float* Uhl = smem;                 // 128*LSTR
  float* Hrl = Uhl + 128 * LSTR;     // 128*LSTR
  float* Hil = Hrl + 128 * LSTR;     // 128*LSTR
  float* Tml = Hil + 128 * LSTR;     // 128*LSTR
  float* av  = Tml + 128 * LSTR;     // 128
  float* bv  = av + 128;             // 128
  float* red = bv + 128;             // 256

  const int m = blockIdx.x >> 4;
  const int b = blockIdx.x & 15;
  const int tid = threadIdx.x;

  const float* Uh = (m == 0) ? Uht : (m == 1) ? Uha : Uhv;
  const float lam = (m == 0) ? LamT[0] : (m == 1) ? LamA[0] : LamV[0];
  const float s = 1.0f / (1.0f + expf(-lam));
  const float os = 1.0f - s;

  // load Uh, init h0 = I/128 (maximally mixed), Hi = 0
  for (int idx = tid; idx < 128 * 128; idx += 256) {
    const int r = idx >> 7, c = idx & 127;
    Uhl[r * LSTR + c] = Uh[idx];
    Hrl[r * LSTR + c] = (r == c) ? (1.0f / 128.0f) : 0.0f;
    Hil[r * LSTR + c] = 0.0f;
  }
  __syncthreads();

  for (int t = 0; t < TT; ++t) {
    const int base = ((m * 16 + b) * TT + t) * 128;
    if (tid < 128) av[tid] = Ag[base + tid];
    else           bv[tid - 128] = Bg[base + (tid - 128)];
    __syncthreads();

    // Hr' = os*(Uh Hr Uh^T) + s*(a a^T + b b^T)
    mm128<0, 0>(Tml, Uhl, Hrl, 1.0f, 0.0f, av, bv, tid);   // Tmp = Uh @ Hr
    __syncthreads();
    mm128<1, 1>(Hrl, Tml, Uhl, os, s, av, bv, tid);        // Hr = os*Tmp@Uh^T + rank2
    __syncthreads();
    // Hi' = os*(Uh Hi Uh^T) + s*(b a^T - a b^T)
    mm128<0, 0>(Tml, Uhl, Hil, 1.0f, 0.0f, av, bv, tid);   // Tmp = Uh @ Hi
    __syncthreads();
    mm128<1, 2>(Hil, Tml, Uhl, os, s, bv, av, tid);        // Hi = os*Tmp@Uh^T + antisym
    __syncthreads();

    // Measurement: p[k] = vr'Hr vr + vi'Hr vi + vi'Hi vr - vr'Hi vi
    float pacc[2];
    for (int eK = 0; eK < 2; ++eK) {
      const float* VTr = eK ? VTrA : VTrE;
      const float* VTi = eK ? VTiA : VTiE;
      const float* Vr  = eK ? VrA  : VrE;
      const float* Vi  = eK ? ViA  : ViE;
      float acc = 0.0f;
      for (int q = 0; q < 4; ++q) {
        const float* Hsrc  = (q < 2) ? Hrl : Hil;
        const float* VTq   = (q & 1) ? VTi : VTr;
        const float* Vrowq = (q == 1 || q == 2) ? Vi : Vr;
        const float sg     = (q == 3) ? -1.0f : 1.0f;
        mm128<2, 0>(Tml, Hsrc, VTq, 1.0f, 0.0f, av, bv, tid);  // Tmp = H @ V^T
        __syncthreads();
        const int k = tid & 127;
        const int hh = tid >> 7;
        const float* vrow = Vrowq + k * 128 + hh * 64;
        float sacc = 0.0f;
        for (int d2 = 0; d2 < 64; ++d2)
          sacc += vrow[d2] * Tml[(hh * 64 + d2) * LSTR + k];
        red[tid] = sacc;
        __syncthreads();
        if (tid < 128) acc += sg * (red[tid] + red[tid + 128]);
        __syncthreads();
      }
      pacc[eK] = acc;
    }
    if (tid < 128) {
      const int idx = b * TT + t;
      Pem[(m * 640 + idx) * 128 + tid] = pacc[0];
      Pam[(m * 640 + idx) * 128 + tid] = pacc[1];
    }
    __syncthreads();
  }
}

// ---------------------------------------------------------------------------
// Kernel 5: QMixture (weighted sum over modalities) + MLP heads + log_softmax.
// grid = 640 blocks of 64 threads.
// ---------------------------------------------------------------------------
__global__ void heads_kernel(const float* __restrict__ Pem,
                             const float* __restrict__ Pam,
                             const float* __restrict__ Wts,
                             const float* __restrict__ W1e, const float* __restrict__ b1e,
                             const float* __restrict__ W2e, const float* __restrict__ b2e,
                             const float* __restrict__ W1a, const float* __restrict__ b1a,
                             const float* __restrict__ W2a, const float* __restrict__ b2a,
                             float* __restrict__ out) {
  const int i = blockIdx.x;   // 0..639
  const int c = threadIdx.x;  // 0..63
  __shared__ float pe_s[128], pa_s[128], h_s[64], o_s[12];
  const float w0 = Wts[i * 3 + 0], w1 = Wts[i * 3 + 1], w2 = Wts[i * 3 + 2];
  for (int d = c; d < 128; d += 64) {
    pe_s[d] = w0 * Pem[(0 * 640 + i) * 128 + d] +
              w1 * Pem[(1 * 640 + i) * 128 + d] +
              w2 * Pem[(2 * 640 + i) * 128 + d];
    pa_s[d] = w0 * Pam[(0 * 640 + i) * 128 + d] +
              w1 * Pam[(1 * 640 + i) * 128 + d] +
              w2 * Pam[(2 * 640 + i) * 128 + d];
  }
  __syncthreads();
  // emotion head
  {
    float hv = b1e[c];
    for (int d = 0; d < 128; ++d) hv += pe_s[d] * W1e[d * 64 + c];
    h_s[c] = fmaxf(hv, 0.0f);
    __syncthreads();
    if (c < 6) {
      float o = b2e[c];
      for (int cc = 0; cc < 64; ++cc) o += h_s[cc] * W2e[cc * 6 + c];
      o_s[c] = tanhf(o);
    }
    __syncthreads();
    if (c == 0) {
      float mx = o_s[0];
      for (int j = 1; j < 6; ++j) mx = fmaxf(mx, o_s[j]);
      float sum = 0.0f;
      for (int j = 0; j < 6; ++j) sum += expf(o_s[j] - mx);
      const float lse = mx + logf(sum);
      for (int j = 0; j < 6; ++j) out[i * 6 + j] = o_s[j] - lse;
    }
    __syncthreads();
  }
  // act head
  {
    float hv = b1a[c];
    for (int d = 0; d < 128; ++d) hv += pa_s[d] * W1a[d * 64 + c];
    h_s[c] = fmaxf(hv, 0.0f);
    __syncthreads();
    if (c < 12) {
      float o = b2a[c];
      for (int cc = 0; cc < 64; ++cc) o += h_s[cc] * W2a[cc * 12 + c];
      o_s[c] = tanhf(o);
    }
    __syncthreads();
    if (c == 0) {
      float mx = o_s[0];
      for (int j = 1; j < 12; ++j) mx = fmaxf(mx, o_s[j]);
      float sum = 0.0f;
      for (int j = 0; j < 12; ++j) sum += expf(o_s[j] - mx);
      const float lse = mx + logf(sum);
      for (int j = 0; j < 12; ++j) out[3840 + i * 12 + j] = o_s[j] - lse;
    }
  }
}

// ---------------------------------------------------------------------------
extern "C" void kernel_launch(void* const* d_in, const int* in_sizes, int n_in,
                              void* d_out, int out_size, void* d_ws, size_t ws_size,
                              hipStream_t stream) {
  (void)in_sizes; (void)n_in; (void)out_size; (void)ws_size;
  const float* x_t   = (const float*)d_in[0];
  const float* x_a   = (const float*)d_in[1];
  const float* x_v   = (const float*)d_in[2];
  const float* smask = (const float*)d_in[3];
  const float* Wp_t = (const float*)d_in[4];
  const float* bp_t = (const float*)d_in[5];
  const float* Ux_t = (const float*)d_in[6];
  const float* Uh_t = (const float*)d_in[7];
  const float* Lm_t = (const float*)d_in[8];
  const float* Wp_a = (const float*)d_in[9];
  const float* bp_a = (const float*)d_in[10];
  const float* Ux_a = (const float*)d_in[11];
  const float* Uh_a = (const float*)d_in[12];
  const float* Lm_a = (const float*)d_in[13];
  const float* Wp_v = (const float*)d_in[14];
  const float* bp_v = (const float*)d_in[15];
  const float* Ux_v = (const float*)d_in[16];
  const float* Uh_v = (const float*)d_in[17];
  const float* Lm_v = (const float*)d_in[18];
  const float* phase_tab = (const float*)d_in[19];
  const float* K_emo = (const float*)d_in[20];
  const float* K_act = (const float*)d_in[21];
  const float* W1e = (const float*)d_in[22];
  const float* b1e = (const float*)d_in[23];
  const float* W2e = (const float*)d_in[24];
  const float* b2e = (const float*)d_in[25];
  const float* W1a = (const float*)d_in[26];
  const float* b1a = (const float*)d_in[27];
  const float* W2a = (const float*)d_in[28];
  const float* b2a = (const float*)d_in[29];

  float* w = (float*)d_ws;
  float* Ag = w;     w += 3 * 640 * 128;
  float* Bg = w;     w += 3 * 640 * 128;
  float* norms = w;  w += 3 * 640;
  float* wts = w;    w += 640 * 3;
  float* VrE = w;    w += 128 * 128;
  float* ViE = w;    w += 128 * 128;
  float* VrA = w;    w += 128 * 128;
  float* ViA = w;    w += 128 * 128;
  float* VTrE = w;   w += 128 * 128;
  float* VTiE = w;   w += 128 * 128;
  float* VTrA = w;   w += 128 * 128;
  float* VTiA = w;   w += 128 * 128;
  float* Pem = w;    w += 3 * 640 * 128;
  float* Pam = w;    w += 3 * 640 * 128;

  prep_kernel<<<3 * 640, 128, 0, stream>>>(
      x_t, x_a, x_v, smask, Wp_t, bp_t, Ux_t, Wp_a, bp_a, Ux_a,
      Wp_v, bp_v, Ux_v, phase_tab, Ag, Bg, norms);
  wts_kernel<<<3, 256, 0, stream>>>(norms, wts);
  normk_kernel<<<256, 128, 0, stream>>>(K_emo, K_act, VrE, ViE, VrA, ViA,
                                        VTrE, VTiE, VTrA, VTiA);
  const size_t shmem = (size_t)(4 * 128 * LSTR + 512) * sizeof(float);
  hipFuncSetAttribute((const void*)qrnn_kernel,
                      hipFuncAttributeMaxDynamicSharedMemorySize, (int)shmem);
  qrnn_kernel<<<48, 256, shmem, stream>>>(
      Uh_t, Uh_a, Uh_v, Lm_t, Lm_a, Lm_v, Ag, Bg,
      VTrE, VTiE, VTrA, VTiA, VrE, ViE, VrA, ViA, Pem, Pam);
  heads_kernel<<<640, 64, 0, stream>>>(Pem, Pam, wts, W1e, b1e, W2e, b2e,
                                       W1a, b1a, W2a, b2a, (float*)d_out);
}